// FeatureInteractionLayer_2680059593352
// MI455X (gfx1250) — compile-verified
//
#include <hip/hip_runtime.h>

// Feature-interaction layer: out[b] = concat(x[b], {x[b,i]*x[b,j] : i<j}), F=128.
// Store-bandwidth bound (~541 MB out @ 23.3 TB/s). Products computed as 16x16
// tiles of the rank-1 outer product via V_WMMA_F32_16X16X4_F32 (K=0 only).
// One row per 32-thread workgroup => uniform SGPR store base (saddr form);
// triu packed offsets are compile-time constants selected per lane-half.

typedef __attribute__((ext_vector_type(2))) float v2f;
typedef __attribute__((ext_vector_type(8))) float v8f;

#define FDIM 128
#define OUTC (FDIM + FDIM * (FDIM - 1) / 2)   // 8256

// packed triu(k=1) index of pair (i,j), j>i, equals triu_g(i) + j
__host__ __device__ constexpr int triu_g(int i) {
  return 126 * i - (i * (i - 1)) / 2 - 1;
}

__global__ __launch_bounds__(32) void feature_interact_wmma(
    const float* __restrict__ x, float* __restrict__ out, int nrows) {
  const int row = blockIdx.x;                  // uniform: one row per wave32
  if (row >= nrows) return;
  const int lane = threadIdx.x;                // 0..31
  const int half = lane >> 4;                  // 0: lanes 0-15, 1: lanes 16-31
  const int lcol = lane & 15;

  const float* __restrict__ xr = x + (size_t)row * FDIM;     // SGPR base
  float* __restrict__ orow = out + (size_t)row * OUTC;       // SGPR base

  // ---- copy x (128 floats, 4 per lane), streaming stores ----
#pragma unroll
  for (int k = 0; k < 4; ++k) {
    float v = xr[k * 32 + lane];
    __builtin_nontemporal_store(v, orow + k * 32 + lane);
  }

  float* __restrict__ oprod = orow + FDIM;     // 8128-element products region

  // Preload the row: chunk t lives in xv[t] on lanes 0-15 (zero on 16-31),
  // matching the K=0 slot of both the A (16x4) and B (4x16) f32 WMMA operands.
  float xv[8];
#pragma unroll
  for (int t = 0; t < 8; ++t) {
    float v = xr[t * 16 + lcol];               // valid address on all lanes
    xv[t] = (half == 0) ? v : 0.0f;
  }

  // 8x8 tile grid over the 128x128 outer product; only tn >= tm needed.
#pragma unroll
  for (int tm = 0; tm < 8; ++tm) {
    v2f a;                                      // A: (M=lcol, K=0) on lanes 0-15
    a.x = xv[tm];
    a.y = 0.0f;

    // 32-bit lane offsets for this tm: element (i = tm*16+r+half*8,
    // j = tn*16+lcol) lives at oprod + idx[r] + tn*16; tn*64B folds into the
    // store immediate. g0/g1 are compile-time constants.
    int idx[8];
#pragma unroll
    for (int r = 0; r < 8; ++r) {
      const int g0 = triu_g(tm * 16 + r);       // constant
      const int g1 = triu_g(tm * 16 + r + 8);   // constant
      idx[r] = lcol + ((half == 0) ? g0 : g1);
    }

#pragma unroll
    for (int tn = tm; tn < 8; ++tn) {
      v2f b;                                    // B: (K=0, N=lcol) on lanes 0-15
      b.x = xv[tn];
      b.y = 0.0f;

      v8f c = {};
      // D = A x B + 0 : 16x16 outer-product tile (v_wmma_f32_16x16x4_f32)
      c = __builtin_amdgcn_wmma_f32_16x16x4_f32(
          /*neg_a=*/false, a, /*neg_b=*/false, b,
          /*c_mod=*/(short)0, c, /*reuse_a=*/false, /*reuse_b=*/false);

#pragma unroll
      for (int r = 0; r < 8; ++r) {
        if (tn > tm) {
          // strictly-upper tile: unconditional streaming store
          __builtin_nontemporal_store(c[r], oprod + idx[r] + tn * 16);
        } else if (lcol > r + half * 8) {
          // diagonal tile: keep only j > i
          __builtin_nontemporal_store(c[r], oprod + idx[r] + tn * 16);
        }
      }
    }
  }
}

extern "C" void kernel_launch(void* const* d_in, const int* in_sizes, int n_in,
                              void* d_out, int out_size, void* d_ws, size_t ws_size,
                              hipStream_t stream) {
  const float* x = (const float*)d_in[0];
  float* out = (float*)d_out;
  const int nrows = in_sizes[0] / FDIM;        // 16384
  feature_interact_wmma<<<nrows, 32, 0, stream>>>(x, out, nrows);
}